// SquareSawVCOLite_31628139167979
// MI455X (gfx1250) — compile-verified
//
#include <hip/hip_runtime.h>
#include <stdint.h>

// ---------------- problem constants ----------------
constexpr int   BS      = 32;
constexpr int   NSAMP   = 262144;
constexpr int   BLOCK   = 256;            // 8 x wave32
constexpr int   ITEMS   = 16;             // contiguous elements per thread
constexpr int   CHUNK   = BLOCK * ITEMS;  // 4096 elements per workgroup tile
constexpr int   NCHUNK  = NSAMP / CHUNK;  // 64 tiles per row
constexpr int   NBLK    = BS * NCHUNK;    // 2048 workgroups
constexpr int   NWAVES  = BLOCK / 32;     // 8

constexpr float TWO_PI_F  = 6.28318530717958647692f;
constexpr float INV_2PI_F = 0.15915494309189533577f;
constexpr float W_SCALE   = TWO_PI_F / 44100.0f;
constexpr float HALF_PI_F = 1.57079632679489661923f;

// LDS tile padded with +1 dword after every 16 dwords (TDM pad feature) so the
// blocked per-thread reads (stride 17 dwords, gcd(17,64)=1) are conflict-free.
constexpr int TILE_LDS = CHUNK + CHUNK / 16 + 8;  // 4360 dwords

// ---------------- gfx1250 feature detection ----------------
#if defined(__HIP_DEVICE_COMPILE__) && defined(__gfx1250__)
#define ON_GFX1250 1
#else
#define ON_GFX1250 0
#endif

#if ON_GFX1250 && __has_builtin(__builtin_amdgcn_wmma_f32_16x16x4_f32)
#define HAVE_WMMA_F32X4 1
#else
#define HAVE_WMMA_F32X4 0
#endif

#if ON_GFX1250 && __has_builtin(__builtin_amdgcn_tensor_load_to_lds) && \
    __has_builtin(__builtin_amdgcn_s_wait_tensorcnt)
#define HAVE_TDM 1
#else
#define HAVE_TDM 0
#endif

#if ON_GFX1250 && __has_builtin(__builtin_amdgcn_tanhf)
#define HAVE_TANH 1
#else
#define HAVE_TANH 0
#endif

typedef __attribute__((ext_vector_type(2))) float v2f;
typedef __attribute__((ext_vector_type(8))) float v8f;

// ---------------- TDM: DMA one padded tile (CHUNK f32) global -> LDS ----------------
#if HAVE_TDM
typedef __attribute__((ext_vector_type(4))) unsigned int v4u_t;
typedef __attribute__((ext_vector_type(8))) int          v8i_t;
typedef __attribute__((ext_vector_type(4))) int          v4i_t;

__device__ __forceinline__ void tdm_load_tile(const float* gsrc, float* lds_dst) {
  const uint64_t ga = (uint64_t)(uintptr_t)gsrc;
  const uint32_t la = (uint32_t)(uintptr_t)lds_dst;  // low 32 bits of generic = LDS offset
  v4u_t g0;
  g0.x = 1u;                                                // count=1, user descriptor
  g0.y = la;                                                // lds_addr
  g0.z = (uint32_t)ga;                                      // global_addr[31:0]
  g0.w = ((uint32_t)(ga >> 32) & 0x01FFFFFFu) | (2u << 30); // global_addr[56:32], type=2
  v8i_t g1;
  // data_size=4B (2<<16), pad_enable (1<<20), pad_interval=3 (16 dw), pad_amount=0 (1 dw)
  g1[0] = (int)((2u << 16) | (1u << 20) | (3u << 22));
  g1[1] = (int)(((uint32_t)CHUNK & 0xFFFFu) << 16);         // tensor_dim0[15:0]
  g1[2] = (int)((uint32_t)CHUNK >> 16);                     // tensor_dim0[31:16], tensor_dim1 lo = 0
  g1[3] = (int)(((uint32_t)CHUNK) << 16);                   // tile_dim0 = CHUNK (1-D tile)
  g1[4] = 0; g1[5] = 0; g1[6] = 0; g1[7] = 0;               // tile_dim1/2=0, strides=0
  v4i_t z4 = {0, 0, 0, 0};
#if __has_include(<hip/amd_detail/amd_gfx1250_TDM.h>)
  v8i_t z8 = {0, 0, 0, 0, 0, 0, 0, 0};
  __builtin_amdgcn_tensor_load_to_lds(g0, g1, z4, z4, z8, 0);  // 6-arg toolchain
#else
  __builtin_amdgcn_tensor_load_to_lds(g0, g1, z4, z4, 0);      // 5-arg ROCm 7.2
#endif
}
#endif  // HAVE_TDM

// ---------------- wave32 inclusive scan of per-lane totals ----------------
// WMMA formulation: A[m][k] = t_k (broadcast down rows), B[k][n] = (k <= n) ? 1 : 0
// => D[m][n] = sum_{k<=n} t_k = inclusive prefix s_n, replicated over all rows m,
// so lane L reads s_{L&15} from VGPR0 of D with no dynamic register indexing.
// K=16 realized as 4 chained V_WMMA_F32_16X16X4_F32 (0/1 matrix is f32-exact).
__device__ __forceinline__ float wave_incl_scan(float tot, int lane) {
#if HAVE_WMMA_F32X4
  const int half = lane >> 4;
  const int col  = lane & 15;
  v8f acc = {0.f, 0.f, 0.f, 0.f, 0.f, 0.f, 0.f, 0.f};
#pragma unroll
  for (int i = 0; i < 4; ++i) {
    // layout (mirrors ISA 16x4 table): VGPR0 holds K=4i+2*half, VGPR1 holds K+1
    const int k0 = 4 * i + 2 * half;
    const int k1 = k0 + 1;
    v2f A, B;
    A.x = __shfl(tot, k0);                 // totals of lanes 0..15
    A.y = __shfl(tot, k1);
    B.x = (k0 <= col) ? 1.0f : 0.0f;
    B.y = (k1 <= col) ? 1.0f : 0.0f;
    acc = __builtin_amdgcn_wmma_f32_16x16x4_f32(false, A, false, B, (short)0, acc,
                                                false, false);
  }
  const float lowTot = __shfl(acc[0], 15); // s_15 = sum of lanes 0..15
  v8f acc2 = {lowTot, lowTot, lowTot, lowTot, lowTot, lowTot, lowTot, lowTot};
#pragma unroll
  for (int i = 0; i < 4; ++i) {
    const int k0 = 4 * i + 2 * half;
    const int k1 = k0 + 1;
    v2f A, B;
    A.x = __shfl(tot, 16 + k0);            // totals of lanes 16..31
    A.y = __shfl(tot, 16 + k1);
    B.x = (k0 <= col) ? 1.0f : 0.0f;
    B.y = (k1 <= col) ? 1.0f : 0.0f;
    acc2 = __builtin_amdgcn_wmma_f32_16x16x4_f32(false, A, false, B, (short)0, acc2,
                                                 false, false);
  }
  return (half == 0) ? acc[0] : acc2[0];
#else
  float incl = tot;
#pragma unroll
  for (int off = 1; off < 32; off <<= 1) {
    float v = __shfl_up(incl, off);
    if (lane >= off) incl += v;
  }
  return incl;
#endif
}

// ---------------- pass 1: per-tile sum(w) and max(f0) ----------------
__global__ __launch_bounds__(BLOCK) void pass_sum(const float* __restrict__ f0,
                                                  float* __restrict__ wsum,
                                                  float* __restrict__ wmax) {
  const int blk = blockIdx.x;
  const float4* p4 = (const float4*)(f0 + (size_t)blk * CHUNK);
  const int t = threadIdx.x;
  float s = 0.f, m = 0.f;
#pragma unroll
  for (int i = 0; i < CHUNK / 4 / BLOCK; ++i) {  // 4 coalesced float4 per thread
    float4 v = p4[t + i * BLOCK];
    s += (v.x + v.y) + (v.z + v.w);
    m = fmaxf(m, fmaxf(fmaxf(v.x, v.y), fmaxf(v.z, v.w)));
  }
#pragma unroll
  for (int off = 16; off > 0; off >>= 1) {
    s += __shfl_down(s, off);
    m = fmaxf(m, __shfl_down(m, off));
  }
  __shared__ float ls[NWAVES], lm[NWAVES];
  if ((t & 31) == 0) { ls[t >> 5] = s; lm[t >> 5] = m; }
  __syncthreads();
  if (t == 0) {
    float S = 0.f, M = 0.f;
#pragma unroll
    for (int w = 0; w < NWAVES; ++w) { S += ls[w]; M = fmaxf(M, lm[w]); }
    wsum[blk] = S * W_SCALE;
    wmax[blk] = M;
  }
}

// ---------------- pass 2: per-row scan of tile sums + n_partials ----------------
__global__ void pass_scan(const float* __restrict__ wsum, const float* __restrict__ wmax,
                          float* __restrict__ woff, float* __restrict__ wnp) {
  const int b = threadIdx.x;
  if (b >= BS) return;
  float run = 0.f, M = 0.f;
  for (int c = 0; c < NCHUNK; ++c) {
    woff[b * NCHUNK + c] = run;
    run += wsum[b * NCHUNK + c];
    M = fmaxf(M, wmax[b * NCHUNK + c]);
  }
  wnp[b] = 12000.0f / (M * log10f(M));
}

// ---------------- pass 3: local scan + oscillator evaluation ----------------
__global__ __launch_bounds__(BLOCK) void pass_vco(const float* __restrict__ f0,
                                                  const float* __restrict__ shp,
                                                  const float* __restrict__ phase,
                                                  const float* __restrict__ woff,
                                                  const float* __restrict__ wnp,
                                                  float* __restrict__ out) {
  const int blk = blockIdx.x;
  const int b   = blk / NCHUNK;
  const size_t gbase = (size_t)blk * CHUNK;  // == b*NSAMP + c*CHUNK
  const float* gf0 = f0 + gbase;

  __shared__ float sf0[TILE_LDS];
  __shared__ float wtotS[NWAVES];

  const int t    = threadIdx.x;
  const int lane = t & 31;
  const int wid  = t >> 5;

#if HAVE_TDM
  if (t < 32) {  // one wave issues the DMA descriptor, waits on TENSORcnt
    tdm_load_tile(gf0, sf0);
    __builtin_amdgcn_s_wait_tensorcnt(0);
  }
  __syncthreads();
#else
  // fallback: coalesced striped loads, padded (stride-17) LDS stores
#pragma unroll
  for (int i = 0; i < ITEMS; ++i) {
    const int e = t + i * BLOCK;
    sf0[e + (e >> 4)] = gf0[e];
  }
  __syncthreads();
#endif

  // shape is consumed elementwise in blocked order: direct 4 x float4 loads
  const float4* gsh4 = (const float4*)(shp + gbase + (size_t)(t * ITEMS));
  float4 sh4[ITEMS / 4];
#pragma unroll
  for (int i = 0; i < ITEMS / 4; ++i) sh4[i] = gsh4[i];

  // per-thread serial inclusive scan over its 16 contiguous elements
  const int rowBase = 17 * t;  // element t*16+i lives at padded index 17*t + i
  float x[ITEMS];
  float run = 0.f;
#pragma unroll
  for (int i = 0; i < ITEMS; ++i) {
    run += sf0[rowBase + i] * W_SCALE;
    x[i] = run;
  }
  const float tot = run;

  // cross-lane scan of thread totals (WMMA), then cross-wave via LDS
  const float incl = wave_incl_scan(tot, lane);
  const float excl = incl - tot;
  if (lane == 31) wtotS[wid] = incl;
  __syncthreads();
  float waveOff = 0.f;
#pragma unroll
  for (int w = 0; w < NWAVES; ++w) waveOff += (w < wid) ? wtotS[w] : 0.f;

  float baseArg = woff[blk] + phase[b] + waveOff + excl;
  // reduce mod 2pi so every per-element arg stays tiny (<= 2pi + ~2.5 rad),
  // safely inside v_sin_f32/v_cos_f32's valid input range
  baseArg -= TWO_PI_F * floorf(baseArg * INV_2PI_F);

  const float kk = HALF_PI_F * wnp[b];  // pi * n_partials / 2
  float4* out4 = (float4*)(out + gbase + (size_t)(t * ITEMS));
#pragma unroll
  for (int q = 0; q < ITEMS / 4; ++q) {
    float r[4];
#pragma unroll
    for (int j = 0; j < 4; ++j) {
      const int i = q * 4 + j;
      const float arg = baseArg + x[i];
      const float sv = __sinf(arg);
      const float cv = __cosf(arg);
#if HAVE_TANH
      const float sq = __builtin_amdgcn_tanhf(kk * sv);
#else
      const float sq = tanhf(kk * sv);
#endif
      const float sh = (j == 0) ? sh4[q].x : (j == 1) ? sh4[q].y : (j == 2) ? sh4[q].z
                                                                            : sh4[q].w;
      r[j] = (1.0f - 0.5f * sh) * sq * (1.0f + sh * cv);
    }
    float4 o;
    o.x = r[0]; o.y = r[1]; o.z = r[2]; o.w = r[3];
    out4[q] = o;  // global_store_b128, 64B-aligned per thread
  }
}

// ---------------- host-side launcher ----------------
extern "C" void kernel_launch(void* const* d_in, const int* in_sizes, int n_in,
                              void* d_out, int out_size, void* d_ws, size_t ws_size,
                              hipStream_t stream) {
  (void)in_sizes; (void)n_in; (void)out_size; (void)ws_size;
  const float* f0    = (const float*)d_in[0];
  const float* shp   = (const float*)d_in[1];
  const float* phase = (const float*)d_in[2];
  float* out = (float*)d_out;

  float* ws   = (float*)d_ws;     // 3*NBLK + BS floats < 26 KB of workspace
  float* wsum = ws;
  float* wmx  = ws + NBLK;
  float* woff = ws + 2 * NBLK;
  float* wnp  = ws + 3 * NBLK;

  pass_sum<<<NBLK, BLOCK, 0, stream>>>(f0, wsum, wmx);
  pass_scan<<<1, 32, 0, stream>>>(wsum, wmx, woff, wnp);
  pass_vco<<<NBLK, BLOCK, 0, stream>>>(f0, shp, phase, woff, wnp, out);
}